// Model_64364379898151
// MI455X (gfx1250) — compile-verified
//
#include <hip/hip_runtime.h>

// out[i] = gen_map[x_gen[i]] + c * clk[i] + d * tdp[i]
// Memory-bound streaming gather+axpy: 16 B/elem, 2 FLOP/elem, AI ~0.125.
// MI455X roofline: 8M elems * 16 B = 134 MB @ 23.3 TB/s -> ~5.8 us. HBM-bound.
// Strategy: NT b128 streaming loads/stores (VMEM), 4KB gather table in LDS
// (ds_load_b32 gathers keep divergent traffic off VMEM), 8-wave blocks.

typedef __attribute__((ext_vector_type(4))) float v4f;
typedef __attribute__((ext_vector_type(4))) int   v4i;

#define NUM_GENS 1024
#define BLOCK    256
#define GRID     2048
#define ITERS    4          // n4 == GRID*BLOCK*ITERS for the reference N

__device__ __forceinline__ void body(unsigned i,
                                     const v4i* __restrict__ g4,
                                     const v4f* __restrict__ clk4,
                                     const v4f* __restrict__ tdp4,
                                     v4f* __restrict__ out4,
                                     const float* __restrict__ lds_map,
                                     float c, float d)
{
    // Streaming inputs: non-temporal b128 loads (no reuse -> TH=NT).
    v4i g = __builtin_nontemporal_load(&g4[i]);
    v4f k = __builtin_nontemporal_load(&clk4[i]);
    v4f t = __builtin_nontemporal_load(&tdp4[i]);

    // LDS gathers (ds_load_b32) from the 4KB bank-resident table.
    float a0 = lds_map[g.x & (NUM_GENS - 1)];
    float a1 = lds_map[g.y & (NUM_GENS - 1)];
    float a2 = lds_map[g.z & (NUM_GENS - 1)];
    float a3 = lds_map[g.w & (NUM_GENS - 1)];

    v4f r;
    r.x = __builtin_fmaf(d, t.x, __builtin_fmaf(c, k.x, a0));
    r.y = __builtin_fmaf(d, t.y, __builtin_fmaf(c, k.y, a1));
    r.z = __builtin_fmaf(d, t.z, __builtin_fmaf(c, k.z, a2));
    r.w = __builtin_fmaf(d, t.w, __builtin_fmaf(c, k.w, a3));

    __builtin_nontemporal_store(r, &out4[i]);
}

__global__ __launch_bounds__(BLOCK) void gather_axpy_kernel(
    const int*   __restrict__ x_gen,
    const float* __restrict__ x_clk,
    const float* __restrict__ x_tdp,
    const float* __restrict__ gen_map,
    const float* __restrict__ c_ptr,
    const float* __restrict__ d_ptr,
    float*       __restrict__ out,
    int n4, int n)
{
    __shared__ float lds_map[NUM_GENS];

    // Stage the 4KB gather table: 256 threads * one float4 = 1024 floats.
    // One global_load_b128 + one ds_store_b128 per thread.
    {
        const v4f* gm4 = (const v4f*)gen_map;
        v4f v = gm4[threadIdx.x];
        ((v4f*)lds_map)[threadIdx.x] = v;
    }

    // Uniform scalars (single-element device arrays) -> s_load.
    const float c = *c_ptr;
    const float d = *d_ptr;

    __syncthreads();

    const v4i* g4   = (const v4i*)x_gen;
    const v4f* clk4 = (const v4f*)x_clk;
    const v4f* tdp4 = (const v4f*)x_tdp;
    v4f*       out4 = (v4f*)out;

    const unsigned tid    = blockIdx.x * BLOCK + threadIdx.x;
    const unsigned stride = GRID * BLOCK;

    if (n4 == (int)(stride * ITERS)) {
        // Fast path for the reference size: fully unrolled, no loop compares,
        // 32-bit offsets so VMEM stays in saddr + scale_offset form.
#pragma unroll
        for (unsigned j = 0; j < ITERS; ++j)
            body(tid + j * stride, g4, clk4, tdp4, out4, lds_map, c, d);
    } else {
        for (unsigned i = tid; i < (unsigned)n4; i += stride)
            body(i, g4, clk4, tdp4, out4, lds_map, c, d);
    }

    // Generic scalar tail (empty for N = 8388608).
    if (blockIdx.x == 0 && threadIdx.x == 0) {
        for (int i = n4 * 4; i < n; ++i) {
            float a = lds_map[x_gen[i] & (NUM_GENS - 1)];
            out[i] = __builtin_fmaf(d, x_tdp[i], __builtin_fmaf(c, x_clk[i], a));
        }
    }
}

extern "C" void kernel_launch(void* const* d_in, const int* in_sizes, int n_in,
                              void* d_out, int out_size, void* d_ws, size_t ws_size,
                              hipStream_t stream) {
    // setup_inputs order:
    // 0: x_gen (int32[N]), 1: x_ix (unused), 2: x_max_clock_speed (f32[N]),
    // 3: x_max_tdp (f32[N]), 4: gen_map (f32[1024]), 5: b (unused),
    // 6: c (f32[1]), 7: d (f32[1])
    const int*   x_gen   = (const int*)d_in[0];
    const float* x_clk   = (const float*)d_in[2];
    const float* x_tdp   = (const float*)d_in[3];
    const float* gen_map = (const float*)d_in[4];
    const float* c_ptr   = (const float*)d_in[6];
    const float* d_ptr   = (const float*)d_in[7];
    float*       out     = (float*)d_out;

    const int n  = in_sizes[0];
    const int n4 = n >> 2;

    gather_axpy_kernel<<<GRID, BLOCK, 0, stream>>>(
        x_gen, x_clk, x_tdp, gen_map, c_ptr, d_ptr, out, n4, n);
}